// TransE_23158463660526
// MI455X (gfx1250) — compile-verified
//
#include <hip/hip_runtime.h>

// TransE multi-tail scoring on MI455X (gfx1250, wave32).
// out[b,n] = -sqrt(||(h_b + r_b) - t_n||^2)
//          = -sqrt(||l_b||^2 + ||t_n||^2 - 2 * l_b . t_n)
// Cross term = 128x10000x256 f32 GEMM -> v_wmma_f32_16x16x4_f32.

typedef __attribute__((ext_vector_type(2))) float v2f;
typedef __attribute__((ext_vector_type(8))) float v8f;
typedef __attribute__((ext_vector_type(4))) float f4;

#define EMB_DIM   256
#define B_ROWS    128
#define NCAND     10000
#define KCHUNK    64
#define LSTRIDE   68   // padded LDS row stride (floats): 16B-aligned + conflict-free column reads

__global__ __launch_bounds__(256) void transe_wmma_kernel(
    const float* __restrict__ ent,      // [100000, 256]
    const float* __restrict__ rel,      // [500, 256]
    const int*  __restrict__ heads,     // [128]
    const int*  __restrict__ rels,      // [128]
    const int*  __restrict__ tails,     // [10000]
    float*      __restrict__ out)       // [128, 10000]
{
    __shared__ float Ls[B_ROWS * LSTRIDE];  // left chunk  [128][68]
    __shared__ float Ts[16 * LSTRIDE];      // tail chunk  [16][68]
    __shared__ float ln[B_ROWS];            // ||h+r||^2 per row
    __shared__ float tn[16];                // ||t||^2 per candidate in tile

    const int tid   = threadIdx.x;
    const int wave  = tid >> 5;             // 0..7 (wave32)
    const int lane  = tid & 31;
    const int tile0 = blockIdx.x * 16;      // first candidate column of this block

    if (tid < B_ROWS) ln[tid] = 0.0f;
    if (tid < 16)     tn[tid] = 0.0f;
    __syncthreads();

    // ---- staging roles ----
    // left: thread -> (row, 32-float half of the 64-wide chunk)
    const int lrow  = tid >> 1;             // 0..127
    const int lhalf = tid & 1;              // 0..1
    const int hidx  = heads[lrow];
    const int ridx  = rels[lrow];
    // tails: thread -> (row 0..15, one float4 of the chunk)
    const int trow  = tid >> 4;             // 0..15
    const int tcol4 = (tid & 15) * 4;
    const int tidx  = tails[tile0 + trow];

    v8f acc = {};  // 16x16 f32 accumulator (8 VGPRs)

    for (int c = 0; c < EMB_DIM / KCHUNK; ++c) {
        const int k0 = c * KCHUNK;

        // --- stage left = h + r (32 floats / thread), accumulate ||l||^2 ---
        {
            const f4* hp  = (const f4*)(ent + (size_t)hidx * EMB_DIM + k0 + lhalf * 32);
            const f4* rp  = (const f4*)(rel + (size_t)ridx * EMB_DIM + k0 + lhalf * 32);
            f4*       dst = (f4*)(Ls + lrow * LSTRIDE + lhalf * 32);
            float part = 0.0f;
#pragma unroll
            for (int j = 0; j < 8; ++j) {
                f4 h4 = hp[j];
                f4 r4 = rp[j];
                f4 l4 = h4 + r4;
                dst[j] = l4;
                part += l4.x * l4.x + l4.y * l4.y + l4.z * l4.z + l4.w * l4.w;
            }
            atomicAdd(&ln[lrow], part);     // ds_add_f32
        }

        // --- stage 16 gathered tail rows (4 floats / thread), accumulate ||t||^2 ---
        {
            f4 t4 = *(const f4*)(ent + (size_t)tidx * EMB_DIM + k0 + tcol4);
            *(f4*)(Ts + trow * LSTRIDE + tcol4) = t4;
            atomicAdd(&tn[trow],
                      t4.x * t4.x + t4.y * t4.y + t4.z * t4.z + t4.w * t4.w);
        }
        __syncthreads();

        // --- 16 x v_wmma_f32_16x16x4_f32 over this K chunk ---
        // A (16x4 f32): lanes 0-15 -> M=0..15 / K={0,1}; lanes 16-31 -> K={2,3}
        // B (4x16 f32): lane n holds column n, same K split.
        const int arow = (wave << 4) + (lane & 15);     // global left row for A
        const int koff = (lane >> 4) << 1;              // 0 or 2
        const float* Abase = Ls + arow * LSTRIDE + koff;
        const float* Bbase = Ts + (lane & 15) * LSTRIDE + koff;
#pragma unroll
        for (int kk = 0; kk < KCHUNK / 4; ++kk) {
            v2f a, b;
            a.x = Abase[kk * 4 + 0];
            a.y = Abase[kk * 4 + 1];
            b.x = Bbase[kk * 4 + 0];
            b.y = Bbase[kk * 4 + 1];
            acc = __builtin_amdgcn_wmma_f32_16x16x4_f32(
                /*neg_a=*/false, a, /*neg_b=*/false, b,
                /*c_mod=*/(short)0, acc, /*reuse_a=*/false, /*reuse_b=*/false);
        }
        __syncthreads();   // protect Ls/Ts before next chunk overwrites
    }

    // --- epilogue: C/D layout (VGPR i: lanes 0-15 -> M=i, lanes 16-31 -> M=i+8) ---
    const int n    = lane & 15;
    const int mhi  = (lane >> 4) << 3;      // 0 or 8
    const float tnv = tn[n];
#pragma unroll
    for (int i = 0; i < 8; ++i) {
        const int grow = (wave << 4) + mhi + i;          // global b row, 0..127
        const float d2 = fmaxf(ln[grow] + tnv - 2.0f * acc[i], 0.0f);
        out[(size_t)grow * NCAND + tile0 + n] = -sqrtf(d2);
    }
}

extern "C" void kernel_launch(void* const* d_in, const int* in_sizes, int n_in,
                              void* d_out, int out_size, void* d_ws, size_t ws_size,
                              hipStream_t stream) {
    const float* ent   = (const float*)d_in[0];   // entity_emb   [100000*256]
    const float* rel   = (const float*)d_in[1];   // relation_emb [500*256]
    const int*   heads = (const int*)d_in[2];     // [128]
    const int*   rels  = (const int*)d_in[3];     // [128]
    const int*   tails = (const int*)d_in[4];     // [10000]
    float*       out   = (float*)d_out;           // [128*10000]

    dim3 grid(NCAND / 16);   // 625 tiles of 16 candidates
    dim3 block(256);         // 8 waves; each wave owns a 16x16 output tile slice
    transe_wmma_kernel<<<grid, block, 0, stream>>>(ent, rel, heads, rels, tails, out);
}